// Head_60224031424789
// MI455X (gfx1250) — compile-verified
//
#include <hip/hip_runtime.h>
#include <hip/hip_bf16.h>
#include <math.h>

// Problem constants (from reference): B=4, T=2048, C=1024, H=128, fp32.
#define BB 4
#define TT 2048
#define CC 1024
#define HH 128

typedef __attribute__((ext_vector_type(2))) float v2f;
typedef __attribute__((ext_vector_type(8))) float v8f;

// D = A(16x4) * B(4x16) + C, full fp32 on the CDNA5 matrix pipeline.
// 8-arg form: (neg_a, A, neg_b, B, c_mod, C, reuse_a, reuse_b)
__device__ __forceinline__ v8f wmma_f32(v2f a, v2f b, v8f c) {
  return __builtin_amdgcn_wmma_f32_16x16x4_f32(false, a, false, b, (short)0, c,
                                               false, false);
}

// ---------------------------------------------------------------------------
// Kernel 1: q/k/v = x @ W{q,k,v}.  One wave computes a 16-row x 128-col slab
// of one projection. A-fragment (x rows) reused across all 8 N-tiles.
// grid = (B*T/16, 3), block = 32 (one wave).
// ---------------------------------------------------------------------------
__global__ __launch_bounds__(32) void qkv_proj(
    const float* __restrict__ x, const float* __restrict__ Wq,
    const float* __restrict__ Wk, const float* __restrict__ Wv,
    float* __restrict__ q, float* __restrict__ k, float* __restrict__ v) {
  const int lane = threadIdx.x & 31;
  const int lo = lane & 15, hi = lane >> 4;
  const int m0 = blockIdx.x * 16;  // row tile within flattened [B*T]

  const float* W;
  float* out;
  if (blockIdx.y == 0)      { W = Wq; out = q; }
  else if (blockIdx.y == 1) { W = Wk; out = k; }
  else                      { W = Wv; out = v; }

  const float* xrow = x + (size_t)(m0 + lo) * CC;
  v8f acc[8] = {};

  for (int k0 = 0; k0 < CC; k0 += 4) {
    v2f a = *(const v2f*)(xrow + k0 + 2 * hi);          // A[m=lo][k0+2hi+{0,1}]
    const float* wr0 = W + (size_t)(k0 + 2 * hi) * HH + lo;
    const float* wr1 = wr0 + HH;
#pragma unroll
    for (int n = 0; n < 8; ++n) {
      v2f bf;
      bf.x = wr0[n * 16];                               // B[k][n=lo]
      bf.y = wr1[n * 16];
      acc[n] = wmma_f32(a, bf, acc[n]);
    }
  }
#pragma unroll
  for (int n = 0; n < 8; ++n)
#pragma unroll
    for (int r = 0; r < 8; ++r)
      out[(size_t)(m0 + r + 8 * hi) * HH + n * 16 + lo] = acc[n][r];
}

// ---------------------------------------------------------------------------
// Kernel 2: S[b,t,s] = scale * q[b,t,:].k[b,s,:], masked (-1e30 where s > t).
// One wave per 16x16 tile; fully-masked tiles (s0 > t0+15) are skipped and
// never read downstream. K = H = 128 -> 32 WMMAs per tile.
// grid = (T/16 [s], T/16 [t], B), block = 32.
// ---------------------------------------------------------------------------
__global__ __launch_bounds__(32) void qk_scores(const float* __restrict__ q,
                                                const float* __restrict__ kmat,
                                                float* __restrict__ S) {
  const int s0 = blockIdx.x * 16;
  const int t0 = blockIdx.y * 16;
  const int b = blockIdx.z;
  if (s0 > t0 + 15) return;  // strictly above the diagonal: never needed

  const int lane = threadIdx.x & 31;
  const int lo = lane & 15, hi = lane >> 4;
  const float* qrow = q + ((size_t)b * TT + t0 + lo) * HH;
  const float* krow = kmat + ((size_t)b * TT + s0 + lo) * HH;

  v8f acc = {};
#pragma unroll
  for (int k0 = 0; k0 < HH; k0 += 4) {
    v2f a = *(const v2f*)(qrow + k0 + 2 * hi);   // A[t=lo][k]
    v2f bf = *(const v2f*)(krow + k0 + 2 * hi);  // B[k][s=lo] = k[s][k]
    acc = wmma_f32(a, bf, acc);
  }

  const float scale = 0.08838834764831845f;  // 1/sqrt(128)
  float* Sout = S + ((size_t)b * TT + t0) * TT + s0;
#pragma unroll
  for (int r = 0; r < 8; ++r) {
    const int t = t0 + r + 8 * hi;
    const int s = s0 + lo;
    const float val = (s <= t) ? acc[r] * scale : -1e30f;
    Sout[(size_t)(r + 8 * hi) * TT + lo] = val;
  }
}

// ---------------------------------------------------------------------------
// Kernel 3: per-COLUMN softmax stats (reference softmaxes over the query axis).
// One thread per (b, s). Outer loop over rows t with lanes on consecutive s
// -> every iteration is a coalesced 128B read that hits L2 (S is L2-resident).
// ---------------------------------------------------------------------------
__global__ void col_softmax_stats(const float* __restrict__ S,
                                  float* __restrict__ colmax,
                                  float* __restrict__ colrcp) {
  const int gid = blockIdx.x * blockDim.x + threadIdx.x;  // b*T + s
  const int b = gid / TT;
  const int s = gid % TT;
  const float* Sb = S + (size_t)b * TT * TT;
  const int tstart = s & ~31;  // warp-uniform: skip all-inactive rows

  float m = -3.0e38f;
  for (int t = tstart; t < TT; ++t)
    if (t >= s) m = fmaxf(m, Sb[(size_t)t * TT + s]);

  float sum = 0.f;
  for (int t = tstart; t < TT; ++t)
    if (t >= s) sum += __expf(Sb[(size_t)t * TT + s] - m);

  colmax[gid] = m;
  colrcp[gid] = 1.0f / sum;
}

// ---------------------------------------------------------------------------
// Kernel 4: out[b,t,:] = sum_s exp(S-colmax[s])*colrcp[s] * v[b,s,:].
// One wave per 16-row slab across all H=128 columns; A-fragment is the
// normalized weight computed on the fly (2 expf per lane per K-step),
// reused across 8 N-tiles. s only runs to t0+16 (causal).
// grid = (T/16, B), block = 32.
// ---------------------------------------------------------------------------
__global__ __launch_bounds__(32) void attn_out(
    const float* __restrict__ S, const float* __restrict__ colmax,
    const float* __restrict__ colrcp, const float* __restrict__ v,
    float* __restrict__ out) {
  const int t0 = blockIdx.x * 16;
  const int b = blockIdx.y;
  const int lane = threadIdx.x & 31;
  const int lo = lane & 15, hi = lane >> 4;

  const float* Srow = S + ((size_t)b * TT + t0 + lo) * TT;
  const float* cm = colmax + (size_t)b * TT;
  const float* cr = colrcp + (size_t)b * TT;
  const float* vb = v + (size_t)b * TT * HH;

  v8f acc[8] = {};
  const int sEnd = t0 + 16;
  for (int s0 = 0; s0 < sEnd; s0 += 4) {
    const int sA = s0 + 2 * hi;
    v2f raw = *(const v2f*)(Srow + sA);
    v2f mx = *(const v2f*)(cm + sA);
    v2f rc = *(const v2f*)(cr + sA);
    v2f a;
    a.x = __expf(raw.x - mx.x) * rc.x;  // masked entries: exp(-1e30-..) -> 0
    a.y = __expf(raw.y - mx.y) * rc.y;

    const float* vr0 = vb + (size_t)sA * HH + lo;
    const float* vr1 = vr0 + HH;
#pragma unroll
    for (int n = 0; n < 8; ++n) {
      v2f bf;
      bf.x = vr0[n * 16];
      bf.y = vr1[n * 16];
      acc[n] = wmma_f32(a, bf, acc[n]);
    }
  }

  float* orow = out + ((size_t)b * TT + t0) * HH;
#pragma unroll
  for (int n = 0; n < 8; ++n)
#pragma unroll
    for (int r = 0; r < 8; ++r)
      orow[(size_t)(r + 8 * hi) * HH + n * 16 + lo] = acc[n][r];
}

// ---------------------------------------------------------------------------
extern "C" void kernel_launch(void* const* d_in, const int* in_sizes, int n_in,
                              void* d_out, int out_size, void* d_ws,
                              size_t ws_size, hipStream_t stream) {
  (void)in_sizes; (void)n_in; (void)out_size; (void)ws_size;
  const float* x = (const float*)d_in[0];
  const float* Wq = (const float*)d_in[1];
  const float* Wk = (const float*)d_in[2];
  const float* Wv = (const float*)d_in[3];
  float* out = (float*)d_out;

  // Workspace layout (floats): q | k | v | S | colmax | colrcp  (~80 MB)
  float* ws = (float*)d_ws;
  float* q = ws;
  float* k = q + (size_t)BB * TT * HH;
  float* v = k + (size_t)BB * TT * HH;
  float* S = v + (size_t)BB * TT * HH;
  float* cmax = S + (size_t)BB * TT * TT;
  float* crcp = cmax + (size_t)BB * TT;

  qkv_proj<<<dim3(BB * TT / 16, 3), 32, 0, stream>>>(x, Wq, Wk, Wv, q, k, v);
  qk_scores<<<dim3(TT / 16, TT / 16, BB), 32, 0, stream>>>(q, k, S);
  col_softmax_stats<<<dim3(BB * TT / 256), 256, 0, stream>>>(S, cmax, crcp);
  attn_out<<<dim3(TT / 16, BB), 32, 0, stream>>>(S, cmax, crcp, v, out);
}